// GCN_EXP_84035330113950
// MI455X (gfx1250) — compile-verified
//
#include <hip/hip_runtime.h>
#include <hip/hip_bf16.h>

#define NNODES 50000
#define NEDGES 800000
#define NGRAPH 32
#define FIN 64
#define HD 128
#define EPSV 1e-5f

typedef __attribute__((ext_vector_type(2))) float v2f;
typedef __attribute__((ext_vector_type(8))) float v8f;

__device__ __forceinline__ void atomAddF(float* p, float v) {
    __hip_atomic_fetch_add(p, v, __ATOMIC_RELAXED, __HIP_MEMORY_SCOPE_AGENT);
}

// ---------------------------------------------------------------- utilities
__global__ void fill_kernel(float* __restrict__ p, float v, int n) {
    int i = blockIdx.x * blockDim.x + threadIdx.x;
    if (i < n) p[i] = v;
}

// deg[dst[e]] += 1  (deg pre-filled with 1.0 for the self loop)
__global__ void degree_kernel(const int* __restrict__ dst, float* __restrict__ deg) {
    int e = blockIdx.x * blockDim.x + threadIdx.x;
    if (e < NEDGES) atomAddF(&deg[dst[e]], 1.0f);
}

__global__ void rsqrt_kernel(float* __restrict__ d) {
    int i = blockIdx.x * blockDim.x + threadIdx.x;
    if (i < NNODES) d[i] = rsqrtf(d[i]);
}

// ------------------------------------------------- WMMA f32 GEMM  H = X @ W
// One wave computes one 16x16 tile of H [NNODES x HD], looping K in steps of 4.
// A 16x4 layout (ISA 7.12.2): lanes 0-15 -> M=lane, K={k0,k0+1};
//                             lanes 16-31 -> M=lane-16, K={k0+2,k0+3}.
// B 4x16 mirrored: lane gives N=lane&15, same K split.
// C/D: VGPR v, lanes 0-15 -> M=v, lanes 16-31 -> M=v+8, N=lane&15.
__global__ void gemm_wmma_kernel(const float* __restrict__ X,
                                 const float* __restrict__ W,
                                 float* __restrict__ Hout, int K) {
    int lane = threadIdx.x & 31;
    int wave = threadIdx.x >> 5;
    int tile = blockIdx.x * 8 + wave;      // 25000 tiles total (exact)
    int tm = tile >> 3;                    // 3125 row tiles
    int tn = tile & 7;                     // 8 col tiles
    int row0 = tm << 4, col0 = tn << 4;
    int mrow  = lane & 15;
    int khalf = (lane >> 4) << 1;          // 0 or 2
    int ncol  = col0 + mrow;

    v8f acc = {};
#if __has_builtin(__builtin_amdgcn_wmma_f32_16x16x4_f32)
    const float* xrow = X + (size_t)(row0 + mrow) * K;
    for (int k0 = 0; k0 < K; k0 += 4) {
        v2f a, b;
        a.x = xrow[k0 + khalf];
        a.y = xrow[k0 + khalf + 1];
        b.x = W[(size_t)(k0 + khalf)     * HD + ncol];
        b.y = W[(size_t)(k0 + khalf + 1) * HD + ncol];
        acc = __builtin_amdgcn_wmma_f32_16x16x4_f32(
                false, a, false, b, (short)0, acc, false, false);
    }
    int mbase = (lane >> 4) << 3;
#else
    int mbase = (lane >> 4) << 3;
    #pragma unroll
    for (int v = 0; v < 8; ++v) {
        float s = 0.f;
        const float* xr = X + (size_t)(row0 + mbase + v) * K;
        for (int k = 0; k < K; ++k) s += xr[k] * W[(size_t)k * HD + ncol];
        acc[v] = s;
    }
#endif
    #pragma unroll
    for (int v = 0; v < 8; ++v)
        Hout[(size_t)(row0 + mbase + v) * HD + ncol] = acc[v];
}

// ----------------------------------------- edge aggregation (L2-resident atomics)
// One wave per edge; each lane handles 4 channels (float4 gather, 4 atomics).
__global__ void edge_agg_kernel(const float* __restrict__ Hm,
                                const int* __restrict__ src,
                                const int* __restrict__ dst,
                                const float* __restrict__ dis,
                                float* __restrict__ agg) {
    unsigned t = blockIdx.x * blockDim.x + threadIdx.x;
    unsigned e = t >> 5;
    if (e >= NEDGES) return;
    int lane = t & 31;
    int s = src[e], d = dst[e];
    float nrm = dis[s] * dis[d];
    const float4* hp = (const float4*)(Hm + (size_t)s * HD);
    float4 hv = hp[lane];
    float* ap = agg + (size_t)d * HD + lane * 4;
    atomAddF(ap + 0, hv.x * nrm);
    atomAddF(ap + 1, hv.y * nrm);
    atomAddF(ap + 2, hv.z * nrm);
    atomAddF(ap + 3, hv.w * nrm);
}

// ------------------- fused: agg += h*dis^2 + bias ; accumulate BN stats
__global__ void selfloop_stats_kernel(float* __restrict__ agg,
                                      const float* __restrict__ Hm,
                                      const float* __restrict__ dis,
                                      const float* __restrict__ bias,
                                      float* __restrict__ sums,
                                      float* __restrict__ sumsq) {
    __shared__ float red[512];
    int c = threadIdx.x & (HD - 1);
    int rhalf = threadIdx.x >> 7;                 // 0/1 : two rows per iter
    float b = bias[c];
    float s = 0.f, s2 = 0.f;
    for (int i = blockIdx.x * 2 + rhalf; i < NNODES; i += gridDim.x * 2) {
        float di = dis[i];
        size_t idx = (size_t)i * HD + c;
        float v = agg[idx] + Hm[idx] * di * di + b;
        agg[idx] = v;
        s += v; s2 += v * v;
    }
    red[threadIdx.x]       = s;
    red[threadIdx.x + 256] = s2;
    __syncthreads();
    if (rhalf == 0) {
        s  += red[threadIdx.x + 128];
        s2 += red[threadIdx.x + 384];
        atomAddF(&sums[c],  s);
        atomAddF(&sumsq[c], s2);
    }
}

__global__ void bn_finalize_kernel(const float* __restrict__ sums,
                                   const float* __restrict__ sumsq,
                                   const float* __restrict__ g,
                                   const float* __restrict__ b,
                                   float* __restrict__ scale,
                                   float* __restrict__ shift, float invn) {
    int c = threadIdx.x;
    if (c < HD) {
        float m  = sums[c] * invn;
        float v  = sumsq[c] * invn - m * m;
        float sc = g[c] * rsqrtf(v + EPSV);
        scale[c] = sc;
        shift[c] = b[c] - m * sc;
    }
}

__global__ void bn_apply_kernel(const float* __restrict__ agg,
                                const float* __restrict__ scale,
                                const float* __restrict__ shift,
                                float* __restrict__ out) {
    int i = blockIdx.x * blockDim.x + threadIdx.x;
    if (i < NNODES * HD) {
        int c = i & (HD - 1);
        out[i] = fmaxf(0.f, agg[i] * scale[c] + shift[c]);
    }
}

// ------------------------------------------------------------ mean pooling
__global__ void pool_kernel(const float* __restrict__ X,
                            const int* __restrict__ batch,
                            float* __restrict__ gsum, float* __restrict__ cnt) {
    unsigned t = blockIdx.x * blockDim.x + threadIdx.x;
    unsigned i = t >> 5;
    if (i >= NNODES) return;
    int lane = t & 31;
    int b = batch[i];
    const float4* xp = (const float4*)(X + (size_t)i * HD);
    float4 xv = xp[lane];
    float* gp = gsum + b * HD + lane * 4;
    atomAddF(gp + 0, xv.x);
    atomAddF(gp + 1, xv.y);
    atomAddF(gp + 2, xv.z);
    atomAddF(gp + 3, xv.w);
    if (lane == 0) atomAddF(&cnt[b], 1.0f);
}

// ---------------------------------------------- FF head, one block, 128 thr
__global__ void head_kernel(const float* __restrict__ gsum,
                            const float* __restrict__ cnt,
                            const float* __restrict__ ffW,
                            const float* __restrict__ ffb,
                            const float* __restrict__ ffbn_g,
                            const float* __restrict__ ffbn_b,
                            const float* __restrict__ linW,
                            const float* __restrict__ linb,
                            float* __restrict__ out) {
    __shared__ float G[NGRAPH * HD];
    int c = threadIdx.x;                       // 0..127
    #pragma unroll
    for (int r = 0; r < NGRAPH; ++r)
        G[r * HD + c] = gsum[r * HD + c] / fmaxf(cnt[r], 1.0f);
    __syncthreads();

    float y[NGRAPH];
    for (int l = 0; l < 2; ++l) {
        const float* Wl = ffW + (size_t)l * HD * HD;
        float bl = ffb[l * HD + c];
        float s = 0.f, s2 = 0.f;
        #pragma unroll
        for (int r = 0; r < NGRAPH; ++r) {
            float acc = bl;
            for (int k = 0; k < HD; ++k) acc += G[r * HD + k] * Wl[(size_t)k * HD + c];
            y[r] = acc; s += acc; s2 += acc * acc;
        }
        float m  = s * (1.0f / NGRAPH);
        float v  = s2 * (1.0f / NGRAPH) - m * m;
        float sc = ffbn_g[l * HD + c] * rsqrtf(v + EPSV);
        float sh = ffbn_b[l * HD + c] - m * sc;
        __syncthreads();
        #pragma unroll
        for (int r = 0; r < NGRAPH; ++r)
            G[r * HD + c] = fmaxf(0.f, y[r] * sc + sh);
        __syncthreads();
    }
    if (c < NGRAPH) {
        float acc = linb[0];
        for (int k = 0; k < HD; ++k) acc += G[c * HD + k] * linW[k];
        out[c] = fmaxf(0.f, acc);
    }
}

// ---------------------------------------------------------------- launcher
extern "C" void kernel_launch(void* const* d_in, const int* in_sizes, int n_in,
                              void* d_out, int out_size, void* d_ws, size_t ws_size,
                              hipStream_t stream) {
    const float* x      = (const float*)d_in[0];
    const int*   ei     = (const int*)  d_in[1];
    const int*   batch  = (const int*)  d_in[2];
    const float* W0     = (const float*)d_in[3];
    const float* b0     = (const float*)d_in[4];
    const float* Wh     = (const float*)d_in[5];
    const float* bh     = (const float*)d_in[6];
    const float* bn_g   = (const float*)d_in[7];
    const float* bn_b   = (const float*)d_in[8];
    const float* ffW    = (const float*)d_in[9];
    const float* ffb    = (const float*)d_in[10];
    const float* ffbn_g = (const float*)d_in[11];
    const float* ffbn_b = (const float*)d_in[12];
    const float* linW   = (const float*)d_in[13];
    const float* linb   = (const float*)d_in[14];
    float* out = (float*)d_out;

    const int* srcIdx = ei;
    const int* dstIdx = ei + NEDGES;

    // workspace carving (256B aligned blocks)
    char* ws = (char*)d_ws;
    size_t off = 0;
    auto carve = [&](size_t nfloats) -> float* {
        float* p = (float*)(ws + off);
        off += (nfloats * sizeof(float) + 255) & ~(size_t)255;
        return p;
    };
    float* dis   = carve(NNODES);
    float* H1    = carve((size_t)NNODES * HD);
    float* H2    = carve((size_t)NNODES * HD);
    float* AGG   = carve((size_t)NNODES * HD);
    float* sums  = carve(HD);   // sums & sumsq contiguous (512B each)
    float* sumsq = carve(HD);
    float* scale = carve(HD);
    float* shift = carve(HD);
    float* gsum  = carve(NGRAPH * HD);  // 16KB, multiple of 256 -> cnt contiguous
    float* cnt   = carve(NGRAPH);
    (void)sumsq; (void)cnt;

    const int BT = 256;
    // degree + inverse sqrt
    fill_kernel  <<<(NNODES + BT - 1) / BT, BT, 0, stream>>>(dis, 1.0f, NNODES);
    degree_kernel<<<(NEDGES + BT - 1) / BT, BT, 0, stream>>>(dstIdx, dis);
    rsqrt_kernel <<<(NNODES + BT - 1) / BT, BT, 0, stream>>>(dis);

    const float* curX = x;
    for (int l = 0; l < 3; ++l) {
        int K = (l == 0) ? FIN : HD;
        const float* Wl = (l == 0) ? W0 : (Wh + (size_t)(l - 1) * HD * HD);
        const float* bl = (l == 0) ? b0 : (bh + (size_t)(l - 1) * HD);
        float* Hbuf = (l & 1) ? H2 : H1;

        fill_kernel<<<(NNODES * HD + BT - 1) / BT, BT, 0, stream>>>(AGG, 0.f, NNODES * HD);
        fill_kernel<<<1, BT, 0, stream>>>(sums, 0.f, 2 * HD);  // sums + sumsq

        gemm_wmma_kernel<<<3125, 256, 0, stream>>>(curX, Wl, Hbuf, K);
        edge_agg_kernel<<<(NEDGES * 32) / BT, BT, 0, stream>>>(Hbuf, srcIdx, dstIdx, dis, AGG);
        selfloop_stats_kernel<<<512, 256, 0, stream>>>(AGG, Hbuf, dis, bl, sums, sumsq);
        bn_finalize_kernel<<<1, 128, 0, stream>>>(sums, sumsq, bn_g + l * HD, bn_b + l * HD,
                                                  scale, shift, 1.0f / NNODES);
        bn_apply_kernel<<<(NNODES * HD + BT - 1) / BT, BT, 0, stream>>>(AGG, scale, shift, Hbuf);
        curX = Hbuf;
    }

    fill_kernel<<<(NGRAPH * HD + NGRAPH + BT - 1) / BT, BT, 0, stream>>>(gsum, 0.f,
                                                                         NGRAPH * HD + NGRAPH);
    pool_kernel<<<(NNODES * 32 + BT - 1) / BT, BT, 0, stream>>>(curX, batch, gsum, cnt);
    head_kernel<<<1, 128, 0, stream>>>(gsum, cnt, ffW, ffb, ffbn_g, ffbn_b, linW, linb, out);
}